// VanillaGNNLayer_53446573032075
// MI455X (gfx1250) — compile-verified
//
#include <hip/hip_runtime.h>
#include <hip/hip_bf16.h>

typedef __attribute__((ext_vector_type(2))) float v2f;
typedef __attribute__((ext_vector_type(8))) float v8f;

#define DIM 128  // DIM_IN == DIM_OUT == 128

// ---------------------------------------------------------------------------
// Kernel 1: h = x @ W using V_WMMA_F32_16X16X4_F32 (exact fp32 path).
// Block = 256 threads = 8 wave32; block computes 128 rows x 128 cols.
// Wave w owns the 16-column tile [16w, 16w+16) for all 8 row-tiles:
//   - its B fragments (16 cols x K=128 of W) live in 64 VGPRs, loaded once
//   - A fragments come from a 64KB LDS stage of the x row-slab
// => W global traffic ~50MB total (vs 400MB for a 16-row blocking),
//    256 WMMAs per wave, B regs reused 8x.
// ---------------------------------------------------------------------------
__global__ __launch_bounds__(256)
void gemm_xw_wmma(const float* __restrict__ x, const float* __restrict__ W,
                  float* __restrict__ h, int n_nodes) {
  __shared__ float sx[128 * DIM];  // 64 KB: 128 rows of x

  const int tid  = threadIdx.x;
  const int wave = tid >> 5;
  const int lane = tid & 31;
  const int half = lane >> 4;   // 0: lanes 0-15, 1: lanes 16-31
  const int r16  = lane & 15;

  const int row0  = blockIdx.x * 128;
  const int nrows = min(128, n_nodes - row0);  // block-uniform
  const int nrt   = (nrows + 15) >> 4;         // live row tiles

  // Stage x[row0 .. row0+128) into LDS (zero-pad tail rows).
  {
    float4* xs = (float4*)sx;
    const int total = (128 * DIM) / 4;  // 4096 float4
    if (nrows == 128) {
      const float4* xg = (const float4*)(x + (size_t)row0 * DIM);
      #pragma unroll 4
      for (int i = tid; i < total; i += 256) xs[i] = xg[i];
    } else {
      for (int i = tid; i < total; i += 256) {
        const int r = i >> 5;  // i / (DIM/4)
        float4 zv = {0.f, 0.f, 0.f, 0.f};
        xs[i] = (r < nrows)
                    ? ((const float4*)(x + (size_t)(row0 + r) * DIM))[i & 31]
                    : zv;
      }
    }
  }

  // Preload this wave's B fragments: W cols [col0, col0+16), all K.
  // B 4x16 layout: row K striped across lanes within each VGPR
  // (lane half h holds rows k0+2h and k0+2h+1 in .x/.y).
  const int col0 = wave * 16;
  v2f bfrag[32];
  #pragma unroll
  for (int kk = 0; kk < 32; ++kk) {
    const int k0 = kk * 4;
    bfrag[kk].x = W[(size_t)(k0 + 2 * half + 0) * DIM + col0 + r16];
    bfrag[kk].y = W[(size_t)(k0 + 2 * half + 1) * DIM + col0 + r16];
  }

  __syncthreads();

  for (int rt = 0; rt < nrt; ++rt) {  // block-uniform trip count
    const int rrow = rt * 16;
    v8f acc = {};
    #pragma unroll
    for (int kk = 0; kk < 32; ++kk) {
      // A 16x4: lane half h holds K = 4kk+2h (v0), 4kk+2h+1 (v1), M = r16
      v2f a = *(const v2f*)&sx[(rrow + r16) * DIM + kk * 4 + 2 * half];
      // 8 args: (neg_a, A, neg_b, B, c_mod, C, reuse_a, reuse_b)
      acc = __builtin_amdgcn_wmma_f32_16x16x4_f32(false, a, false, bfrag[kk],
                                                  (short)0, acc, false, false);
    }
    // D layout: VGPR i -> row (row0 + rrow + 8*half + i), col (col0 + r16)
    const int rbase = row0 + rrow + 8 * half;
    if (rbase + 8 <= n_nodes) {
      float* hp = h + (size_t)rbase * DIM + col0 + r16;
      #pragma unroll
      for (int i = 0; i < 8; ++i) hp[(size_t)i * DIM] = acc[i];
    } else {
      #pragma unroll
      for (int i = 0; i < 8; ++i)
        if (rbase + i < n_nodes) h[(size_t)(rbase + i) * DIM + col0 + r16] = acc[i];
    }
  }
}

// ---------------------------------------------------------------------------
// Kernel 2: COO SpMM. One wave32 per edge. Edge index is forced into an SGPR
// via readfirstlane so rows/cols/vals come through scalar loads (constant
// cache), the h-row gather is one coalesced 512B float4 load per wave, and
// accumulation is 4 non-returning global_atomic_add_f32 per lane (STOREcnt
// only; h/out are resident in the 192MB L2).
// ---------------------------------------------------------------------------
__global__ __launch_bounds__(256)
void spmm_coo_atomic(const float* __restrict__ h, const float* __restrict__ vals,
                     const int* __restrict__ rows, const int* __restrict__ cols,
                     float* __restrict__ out, int n_edges) {
  const int wlocal = __builtin_amdgcn_readfirstlane((int)(threadIdx.x >> 5));
  const int gwave  = (int)blockIdx.x * 8 + wlocal;  // wave-uniform (SGPR)
  const int lane   = threadIdx.x & 31;
  if (gwave >= n_edges) return;

  const int   c = cols[gwave];   // scalar loads
  const int   r = rows[gwave];
  const float v = vals[gwave];

  const float4 m = ((const float4*)(h + (size_t)c * DIM))[lane];

  float* op = out + (size_t)r * DIM + (lane << 2);
  atomicAdd(op + 0, v * m.x);
  atomicAdd(op + 1, v * m.y);
  atomicAdd(op + 2, v * m.z);
  atomicAdd(op + 3, v * m.w);
}

// ---------------------------------------------------------------------------
extern "C" void kernel_launch(void* const* d_in, const int* in_sizes, int n_in,
                              void* d_out, int out_size, void* d_ws, size_t ws_size,
                              hipStream_t stream) {
  const float* x    = (const float*)d_in[0];
  const float* W    = (const float*)d_in[1];
  const float* vals = (const float*)d_in[2];
  const int*   rows = (const int*)d_in[3];
  const int*   cols = (const int*)d_in[4];

  const int n_nodes = in_sizes[0] / DIM;
  const int n_edges = in_sizes[2];

  float* out = (float*)d_out;
  float* h   = (float*)d_ws;  // 100000*128*4 = 51.2 MB scratch

  // out accumulates via atomics -> must start at zero (capture-safe memset)
  hipMemsetAsync(d_out, 0, (size_t)out_size * sizeof(float), stream);

  const int row_blocks = (n_nodes + 127) / 128;  // 782 for N=100000
  gemm_xw_wmma<<<row_blocks, 256, 0, stream>>>(x, W, h, n_nodes);

  const int edges_per_block = 256 / 32;  // 8 waves per block
  const int blocks = (n_edges + edges_per_block - 1) / edges_per_block;
  spmm_coo_atomic<<<blocks, 256, 0, stream>>>(h, vals, rows, cols, out, n_edges);
}